// XCentericLoss_64802466562637
// MI455X (gfx1250) — compile-verified
//
#include <hip/hip_runtime.h>
#include <math.h>
#include <stdint.h>

// ---------------- problem constants (from reference) ----------------
#define NPTS   4096
#define DIM    256
#define NBATCH 4          // batches 1..4 of T=5
#define REG    2.0f       // SINKHORN_REG
#define ITERS  100        // SINKHORN_ITERS
#define TOPK   5
#define HINGE_V 0.1f
#define LAM_DEN 5.0f
#define LMU   (-8.317766166719343f)   // -log(4096)
#define LNU   (-8.317766166719343f)

typedef __attribute__((ext_vector_type(2))) float v2f;
typedef __attribute__((ext_vector_type(8))) float v8f;

// ---------------- CDNA5 async global->LDS copy (ASYNCcnt-tracked) ----------------
// Writes 16 bytes straight into LDS with no VGPR landing zone.
__device__ __forceinline__ void async_copy_b128(void* lds_ptr, const float* gptr) {
    const uint32_t lds_addr = (uint32_t)(uintptr_t)lds_ptr; // flat addr low 32 bits = LDS byte offset
    asm volatile("global_load_async_to_lds_b128 %0, %1, off"
                 :: "v"(lds_addr), "v"(gptr)
                 : "memory");
}

// ---------------- row squared-norms: aa[i] = sum_k X[i][k]^2 ----------------
__global__ void row_norms_kernel(const float* __restrict__ X, float* __restrict__ out) {
    __shared__ float red[64];
    const int row = blockIdx.x;
    const float* p = X + (size_t)row * DIM;
    float s = 0.f;
    for (int k = threadIdx.x; k < DIM; k += 64) { float v = p[k]; s += v * v; }
    red[threadIdx.x] = s;
    __syncthreads();
    for (int off = 32; off > 0; off >>= 1) {
        if (threadIdx.x < off) red[threadIdx.x] += red[threadIdx.x + off];
        __syncthreads();
    }
    if (threadIdx.x == 0) out[row] = red[0];
}

// ---------------- WMMA f32 GEMM + fused distance epilogue ----------------
// Block: 256 threads (8 waves), computes a 128x128 tile of lK.
// Wave tile: 64(M) x 32(N) -> 4x2 accumulators of V_WMMA_F32_16X16X4_F32.
// Double-buffered LDS chunks (K=16) filled by GLOBAL_LOAD_ASYNC_TO_LDS_B128;
// row stride padded to 20 floats -> conflict-free ds_load_2addr_b64 fragment
// reads and 16B-aligned async stores.
__global__ __launch_bounds__(256) void gemm_lk_kernel(
    const float* __restrict__ Xp, const float* __restrict__ Xt,
    const float* __restrict__ aa, const float* __restrict__ bb,
    float* __restrict__ lK) {
    __shared__ float As[2][128][20];
    __shared__ float Bs[2][128][20];

    const int t     = threadIdx.x;
    const int lane  = t & 31;
    const int w     = t >> 5;
    const int iBase = blockIdx.x * 128;
    const int jBase = blockIdx.y * 128;
    const int mOff  = (w >> 2) * 64;   // 2 waves along M
    const int nOff  = (w & 3) * 32;    // 4 waves along N
    const int r     = lane & 15;       // row (A) / col (B) within 16
    const int khalf = (lane >> 4) * 2; // K pair select per ISA layout

    // per-thread staging assignment: 2 x (row, float4-col) for A and B each
    const int row0 = t >> 2;            // idx = t      -> rows   0..63
    const int row1 = (256 + t) >> 2;    // idx = 256+t  -> rows  64..127
    const int c40  = t & 3;
    const int c41  = (256 + t) & 3;

    v8f acc[4][2];
    const v8f vz = {0.f,0.f,0.f,0.f,0.f,0.f,0.f,0.f};
    #pragma unroll
    for (int mt = 0; mt < 4; ++mt)
        #pragma unroll
        for (int nt = 0; nt < 2; ++nt) acc[mt][nt] = vz;

    // prefetch chunk 0 into buffer 0 (4 async b128 instructions per wave)
    {
        const int k0 = 0;
        async_copy_b128(&As[0][row0][c40 * 4], Xp + (size_t)(iBase + row0) * DIM + k0 + c40 * 4);
        async_copy_b128(&Bs[0][row0][c40 * 4], Xt + (size_t)(jBase + row0) * DIM + k0 + c40 * 4);
        async_copy_b128(&As[0][row1][c41 * 4], Xp + (size_t)(iBase + row1) * DIM + k0 + c41 * 4);
        async_copy_b128(&Bs[0][row1][c41 * 4], Xt + (size_t)(jBase + row1) * DIM + k0 + c41 * 4);
    }

    for (int c = 0; c < DIM / 16; ++c) {
        const int buf = c & 1;
        if (c + 1 < DIM / 16) {
            __syncthreads();  // all waves done computing from buf^1 -> safe to overwrite
            const int k0 = (c + 1) * 16;
            const int nb = buf ^ 1;
            async_copy_b128(&As[nb][row0][c40 * 4], Xp + (size_t)(iBase + row0) * DIM + k0 + c40 * 4);
            async_copy_b128(&Bs[nb][row0][c40 * 4], Xt + (size_t)(jBase + row0) * DIM + k0 + c40 * 4);
            async_copy_b128(&As[nb][row1][c41 * 4], Xp + (size_t)(iBase + row1) * DIM + k0 + c41 * 4);
            async_copy_b128(&Bs[nb][row1][c41 * 4], Xt + (size_t)(jBase + row1) * DIM + k0 + c41 * 4);
            // chunk c (4 older async ops) retired; chunk c+1 (4 ops) may stay in flight
            asm volatile("s_wait_asynccnt 0x4" ::: "memory");
        } else {
            asm volatile("s_wait_asynccnt 0x0" ::: "memory");
        }
        __syncthreads();  // publish chunk c to all waves

        #pragma unroll
        for (int kk = 0; kk < 16; kk += 4) {
            const int ks = kk + khalf;
            v2f a[4], bf[2];
            #pragma unroll
            for (int mt = 0; mt < 4; ++mt) {
                a[mt].x = As[buf][mOff + mt * 16 + r][ks];
                a[mt].y = As[buf][mOff + mt * 16 + r][ks + 1];
            }
            #pragma unroll
            for (int nt = 0; nt < 2; ++nt) {
                bf[nt].x = Bs[buf][nOff + nt * 16 + r][ks];
                bf[nt].y = Bs[buf][nOff + nt * 16 + r][ks + 1];
            }
            #pragma unroll
            for (int mt = 0; mt < 4; ++mt)
                #pragma unroll
                for (int nt = 0; nt < 2; ++nt)
                    acc[mt][nt] = __builtin_amdgcn_wmma_f32_16x16x4_f32(
                        false, a[mt], false, bf[nt], (short)0, acc[mt][nt], false, false);
        }
    }

    // epilogue: lK = -max(aa_i + bb_j - 2*dot, 0) / REG
    const int rowSel = (lane >> 4) * 8;  // C/D layout: M = vgpr + 8*lane[4]
    #pragma unroll
    for (int mt = 0; mt < 4; ++mt) {
        #pragma unroll
        for (int nt = 0; nt < 2; ++nt) {
            const int gj  = jBase + nOff + nt * 16 + r;
            const float bbv = bb[gj];
            #pragma unroll
            for (int v = 0; v < 8; ++v) {
                const int gi = iBase + mOff + mt * 16 + rowSel + v;
                const float cc = acc[mt][nt][v];
                const float sq = fmaxf(aa[gi] + bbv - 2.f * cc, 0.f);
                lK[(size_t)gi * NPTS + gj] = sq * (-1.0f / REG);
            }
        }
    }
}

// ---------------- density: per-row 5 smallest sq, hinge, accumulate ----------------
__global__ __launch_bounds__(128) void density_kernel(const float* __restrict__ lK,
                                                      float* __restrict__ accum) {
    __shared__ float cand[128 * TOPK];
    const int row = blockIdx.x;
    const float* p = lK + (size_t)row * NPTS;
    float best[TOPK];
    #pragma unroll
    for (int q = 0; q < TOPK; ++q) best[q] = 3.4e38f;
    for (int j = threadIdx.x; j < NPTS; j += 128) {
        const float sq = -REG * p[j];
        int mi = 0; float mv = best[0];
        #pragma unroll
        for (int q = 1; q < TOPK; ++q) if (best[q] > mv) { mv = best[q]; mi = q; }
        if (sq < mv) best[mi] = sq;
    }
    #pragma unroll
    for (int q = 0; q < TOPK; ++q) cand[threadIdx.x * TOPK + q] = best[q];
    __syncthreads();
    if (threadIdx.x == 0) {
        float b2[TOPK];
        #pragma unroll
        for (int q = 0; q < TOPK; ++q) b2[q] = 3.4e38f;
        for (int c = 0; c < 128 * TOPK; ++c) {
            const float x = cand[c];
            int mi = 0; float mv = b2[0];
            #pragma unroll
            for (int q = 1; q < TOPK; ++q) if (b2[q] > mv) { mv = b2[q]; mi = q; }
            if (x < mv) b2[mi] = x;
        }
        float s = 0.f;
        #pragma unroll
        for (int q = 0; q < TOPK; ++q) s += fmaxf(sqrtf(b2[q]) - HINGE_V, 0.f);
        atomicAdd(accum + 1, s);
    }
}

// ---------------- sinkhorn: row logsumexp -> la ----------------
__global__ __launch_bounds__(256) void row_lse_kernel(const float* __restrict__ lK,
                                                      const float* __restrict__ lb,
                                                      float* __restrict__ la) {
    __shared__ float sm[256], ss[256];
    const int row = blockIdx.x;
    const float* p = lK + (size_t)row * NPTS;
    float m = -3.4e38f, s = 0.f;
    for (int j = threadIdx.x; j < NPTS; j += 256) {
        const float x  = p[j] + lb[j];
        const float nm = fmaxf(m, x);
        s = s * __expf(m - nm) + __expf(x - nm);
        m = nm;
    }
    sm[threadIdx.x] = m; ss[threadIdx.x] = s;
    __syncthreads();
    for (int off = 128; off > 0; off >>= 1) {
        if (threadIdx.x < off) {
            const float m1 = sm[threadIdx.x],       s1 = ss[threadIdx.x];
            const float m2 = sm[threadIdx.x + off], s2 = ss[threadIdx.x + off];
            const float nm = fmaxf(m1, m2);
            sm[threadIdx.x] = nm;
            ss[threadIdx.x] = s1 * __expf(m1 - nm) + s2 * __expf(m2 - nm);
        }
        __syncthreads();
    }
    if (threadIdx.x == 0) la[row] = LMU - (sm[0] + __logf(ss[0]));
}

// ---------------- sinkhorn: column logsumexp partials (32 i-chunks of 128 rows) ----------------
__global__ __launch_bounds__(256) void col_lse_partial_kernel(const float* __restrict__ lK,
                                                              const float* __restrict__ la,
                                                              float2* __restrict__ part) {
    const int j  = blockIdx.x * 256 + threadIdx.x;
    const int i0 = blockIdx.y * 128;
    float m = -3.4e38f, s = 0.f;
    const float* p = lK + (size_t)i0 * NPTS + j;
    for (int i = 0; i < 128; ++i) {
        const float x  = p[(size_t)i * NPTS] + la[i0 + i];
        const float nm = fmaxf(m, x);
        s = s * __expf(m - nm) + __expf(x - nm);
        m = nm;
    }
    part[blockIdx.y * NPTS + j] = make_float2(m, s);
}

__global__ __launch_bounds__(256) void col_lse_combine_kernel(const float2* __restrict__ part,
                                                              float* __restrict__ lb) {
    const int j = blockIdx.x * 256 + threadIdx.x;
    float m = -3.4e38f, s = 0.f;
    for (int c = 0; c < 32; ++c) {
        const float2 v = part[c * NPTS + j];
        const float nm = fmaxf(m, v.x);
        s = s * __expf(m - nm) + v.y * __expf(v.x - nm);
        m = nm;
    }
    lb[j] = LNU - (m + __logf(s));
}

// ---------------- ot loss: sum exp(la+lK+lb) * (-REG*lK) ----------------
__global__ __launch_bounds__(256) void ot_kernel(const float* __restrict__ lK,
                                                 const float* __restrict__ la,
                                                 const float* __restrict__ lb,
                                                 float* __restrict__ accum) {
    __shared__ float red[256];
    float local = 0.f;
    const size_t total = (size_t)NPTS * NPTS;
    for (size_t idx = (size_t)blockIdx.x * 256 + threadIdx.x; idx < total;
         idx += (size_t)gridDim.x * 256) {
        const int i = (int)(idx >> 12);
        const int j = (int)(idx & (NPTS - 1));
        const float v = lK[idx];
        const float p = __expf(la[i] + v + lb[j]);
        local += p * (-REG * v);
    }
    red[threadIdx.x] = local;
    __syncthreads();
    for (int off = 128; off > 0; off >>= 1) {
        if (threadIdx.x < off) red[threadIdx.x] += red[threadIdx.x + off];
        __syncthreads();
    }
    if (threadIdx.x == 0) atomicAdd(accum, red[0]);
}

__global__ void zero_kernel(float* __restrict__ p, int n) {
    const int i = blockIdx.x * blockDim.x + threadIdx.x;
    if (i < n) p[i] = 0.f;
}

__global__ void finalize_kernel(const float* __restrict__ accum, float* __restrict__ out) {
    if (threadIdx.x == 0 && blockIdx.x == 0)
        out[0] = accum[0] + LAM_DEN * accum[1] / (float)(NBATCH * NPTS * TOPK);
}

// ---------------- driver ----------------
extern "C" void kernel_launch(void* const* d_in, const int* in_sizes, int n_in,
                              void* d_out, int out_size, void* d_ws, size_t ws_size,
                              hipStream_t stream) {
    const float* X_pred = (const float*)d_in[0];
    const float* X      = (const float*)d_in[1];
    float* ws = (float*)d_ws;

    // workspace layout (floats): one batch processed at a time, lK is L2-resident (64MB << 192MB L2)
    float*  lK    = ws;                               // 4096*4096
    float*  aa    = ws + (size_t)NPTS * NPTS;         // 4096
    float*  bb    = aa + NPTS;                        // 4096
    float*  la    = bb + NPTS;                        // 4096
    float*  lb    = la + NPTS;                        // 4096 (contiguous with la)
    float2* part  = (float2*)(lb + NPTS);             // 32*4096 float2
    float*  accum = (float*)(part + 32 * NPTS);       // [ot_sum, density_sum]

    zero_kernel<<<1, 64, 0, stream>>>(accum, 2);

    for (int b = 1; b <= NBATCH; ++b) {
        const float* Xp = X_pred + (size_t)b * NPTS * DIM;
        const float* Xt = X      + (size_t)b * NPTS * DIM;

        row_norms_kernel<<<NPTS, 64, 0, stream>>>(Xp, aa);
        row_norms_kernel<<<NPTS, 64, 0, stream>>>(Xt, bb);

        dim3 gg(NPTS / 128, NPTS / 128);
        gemm_lk_kernel<<<gg, 256, 0, stream>>>(Xp, Xt, aa, bb, lK);

        density_kernel<<<NPTS, 128, 0, stream>>>(lK, accum);

        zero_kernel<<<(2 * NPTS + 255) / 256, 256, 0, stream>>>(la, 2 * NPTS);
        for (int it = 0; it < ITERS; ++it) {
            row_lse_kernel<<<NPTS, 256, 0, stream>>>(lK, lb, la);
            col_lse_partial_kernel<<<dim3(16, 32), 256, 0, stream>>>(lK, la, part);
            col_lse_combine_kernel<<<16, 256, 0, stream>>>(part, lb);
        }

        ot_kernel<<<4096, 256, 0, stream>>>(lK, la, lb, accum);
    }

    finalize_kernel<<<1, 64, 0, stream>>>(accum, (float*)d_out);
}